// SparseLDS_41927470744182
// MI455X (gfx1250) — compile-verified
//
#include <hip/hip_runtime.h>

typedef float v2f __attribute__((ext_vector_type(2)));
typedef float v8f __attribute__((ext_vector_type(8)));

#define T_LEN 512
#define B_DIM 16
#define K_DIM 64
#define N_DIM 64
#define M_DIM 128
#define P_DIM 512   // padded power axis: p = 510-u for u=0..510, slot 511 = 0

// workspace layout (float offsets)
#define WS_ANG 0
#define WS_BPR (WS_ANG + K_DIM*N_DIM)
#define WS_BPI (WS_BPR + K_DIM*N_DIM)
#define WS_XS  (WS_BPI + K_DIM*N_DIM)                 // [K][B][P] staged input
#define WS_STR (WS_XS  + K_DIM*B_DIM*P_DIM)
#define WS_STI (WS_STR + B_DIM*K_DIM*N_DIM)
#define WS_CSR (WS_STI + B_DIM*K_DIM*N_DIM)
#define WS_CSI (WS_CSR + B_DIM*K_DIM*M_DIM)

// ---------------- kernel 0: stage x into power-ordered transposed padded layout -------
// xs[k][b][p] = x[510-p, b, k]  (p=0..510), xs[k][b][511] = 0
// coalesced reads of x; one-shot 2MB staging so scattered writes are fine.
__global__ void lds_stage_x(const float* __restrict__ x, float* __restrict__ ws) {
    int tid = blockIdx.x * blockDim.x + threadIdx.x;           // tid = u*1024 + b*64 + k
    if (tid >= K_DIM * B_DIM * P_DIM) return;
    int u = tid >> 10;
    int r = tid & 1023;
    int b = r >> 6, k = r & 63;
    float v = (u <= 510) ? x[tid] : 0.f;
    int p = (u <= 510) ? (510 - u) : 511;
    ws[WS_XS + k * (B_DIM * P_DIM) + b * P_DIM + p] = v;
}

// ---------------- kernel 1: lambda angles + B' ----------------
// Bp[k,j] = exp(-sum_{i!=j} log(1 - lam_i/lam_j)), lam = e^{i*ang}
__global__ void lds_lam_bp(const float* __restrict__ theta, float* __restrict__ ws) {
    int tid = blockIdx.x * blockDim.x + threadIdx.x;
    if (tid >= K_DIM * N_DIM) return;
    int k = tid >> 6, j = tid & 63;
    float thj = theta[k * 32 + (j & 31)];
    float angj = (j < 32) ? thj : -thj;
    float Sr = 0.f, Si = 0.f;
    for (int i = 0; i < N_DIM; ++i) {
        if (i == j) continue;
        float thi = theta[k * 32 + (i & 31)];
        float angi = (i < 32) ? thi : -thi;
        float d = angi - angj;              // lam_i/lam_j = e^{i d}
        float sd, cd; __sincosf(d, &sd, &cd);
        float zr = 1.f - cd, zi = -sd;      // 1 - e^{i d}
        Sr += 0.5f * __logf(zr * zr + zi * zi);
        Si += atan2f(zi, zr);
    }
    float er = __expf(-Sr);
    float sb, cb; __sincosf(-Si, &sb, &cb);
    ws[WS_ANG + tid] = angj;
    ws[WS_BPR + tid] = er * cb;
    ws[WS_BPI + tid] = er * sb;
}

// ---------------- kernel 2: state via GEMM over time ----------------
// sT[b,k,n] = Bp[k,n] * sum_p e^{i*p*ang[k,n]} * xs[k][b][p]
// one wave per (k, 16-wide n-tile); WMMA f32 16x16x4, K-dim = power axis.
// Guard-free inner loop: pad slot's zero data kills the p=511 term.
__global__ void __launch_bounds__(32) lds_state(float* __restrict__ ws) {
    int k   = blockIdx.x >> 2;
    int nb  = (blockIdx.x & 3) * 16;
    int lane = threadIdx.x;
    int half = lane >> 4, l16 = lane & 15;
    int n = nb + l16;
    float th = ws[WS_ANG + k * N_DIM + n];
    const float* xrow = ws + WS_XS + k * (B_DIM * P_DIM) + l16 * P_DIM;

    v8f cr = {}; v8f ci = {};
    for (int p0 = 0; p0 < P_DIM; p0 += 4) {
        int pa = p0 + 2 * half;                      // this half-wave's two K-slots
        // A: rows = batch (lane 0..15), cols = power chunk; 64-bit aligned load
        v2f a = *(const v2f*)(xrow + pa);
        // B: rows = power chunk, cols = n (per lane); W = e^{i p th}
        float s0, c0, s1, c1;
        __sincosf((float)pa * th, &s0, &c0);
        __sincosf((float)(pa + 1) * th, &s1, &c1);
        v2f br, bi;
        br.x = c0; br.y = c1;
        bi.x = s0; bi.y = s1;
        cr = __builtin_amdgcn_wmma_f32_16x16x4_f32(false, a, false, br, (short)0, cr, false, false);
        ci = __builtin_amdgcn_wmma_f32_16x16x4_f32(false, a, false, bi, (short)0, ci, false, false);
    }
    float bpr = ws[WS_BPR + k * N_DIM + n];
    float bpi = ws[WS_BPI + k * N_DIM + n];
#pragma unroll
    for (int v = 0; v < 8; ++v) {
        int b = v + 8 * half;                        // C/D layout: M = vgpr + 8*(lane/16)
        int idx = b * (K_DIM * N_DIM) + k * N_DIM + n;
        float r = cr[v], im = ci[v];
        ws[WS_STR + idx] = r * bpr - im * bpi;       // complex multiply by B'
        ws[WS_STI + idx] = r * bpi + im * bpr;
    }
}

// ---------------- kernel 3: output projection ----------------
// Cs[b,k,m] = sum_n sT[b,k,n] * C[k,m,n] + x[510,b,k]*D[k,m],  C = e^{lnC}
// one wave per (k, 16-wide m-tile); complex GEMM = 4 real WMMAs per K-chunk
__global__ void __launch_bounds__(32) lds_proj(const float* __restrict__ x,
                                               const float* __restrict__ lnCr,
                                               const float* __restrict__ lnCi,
                                               const float* __restrict__ Dm,
                                               float* __restrict__ ws) {
    int k   = blockIdx.x >> 3;
    int mb  = (blockIdx.x & 7) * 16;
    int lane = threadIdx.x;
    int half = lane >> 4, l16 = lane & 15;
    int m = mb + l16;
    const float* strow = ws + WS_STR + l16 * (K_DIM * N_DIM) + k * N_DIM;
    const float* sirow = ws + WS_STI + l16 * (K_DIM * N_DIM) + k * N_DIM;
    const float* crow_r = lnCr + k * (M_DIM * N_DIM) + m * N_DIM;
    const float* crow_i = lnCi + k * (M_DIM * N_DIM) + m * N_DIM;

    v8f accR = {}; v8f accI = {};
    for (int n0 = 0; n0 < N_DIM; n0 += 4) {
        int na = n0 + 2 * half;                      // even -> 8B aligned v2f loads
        v2f ar = *(const v2f*)(strow + na);
        v2f ai = *(const v2f*)(sirow + na);
        v2f lr = *(const v2f*)(crow_r + na);
        v2f li = *(const v2f*)(crow_i + na);
        float e0 = __expf(lr.x), e1 = __expf(lr.y);
        float s0, c0, s1, c1;
        __sincosf(li.x, &s0, &c0);
        __sincosf(li.y, &s1, &c1);
        v2f bcr, bci, bciN;
        bcr.x  = e0 * c0;  bcr.y  = e1 * c1;
        bci.x  = e0 * s0;  bci.y  = e1 * s1;
        bciN.x = -bci.x;   bciN.y = -bci.y;  // f32 WMMA NEG only allows C-negate -> negate operand
        accR = __builtin_amdgcn_wmma_f32_16x16x4_f32(false, ar, false, bcr,  (short)0, accR, false, false);
        accR = __builtin_amdgcn_wmma_f32_16x16x4_f32(false, ai, false, bciN, (short)0, accR, false, false);
        accI = __builtin_amdgcn_wmma_f32_16x16x4_f32(false, ar, false, bci,  (short)0, accI, false, false);
        accI = __builtin_amdgcn_wmma_f32_16x16x4_f32(false, ai, false, bcr,  (short)0, accI, false, false);
    }
    float dkm = Dm[k * M_DIM + m];
#pragma unroll
    for (int v = 0; v < 8; ++v) {
        int b = v + 8 * half;
        float x510 = x[510 * (B_DIM * K_DIM) + b * K_DIM + k];   // xc[-1] = x[T-2]
        int idx = b * (K_DIM * M_DIM) + k * M_DIM + m;
        ws[WS_CSR + idx] = accR[v] + x510 * dkm;     // D and x real -> imag untouched
        ws[WS_CSI + idx] = accI[v];
    }
}

// ---------------- kernel 4: mean over k + Do, interleaved complex out ----------------
__global__ void lds_reduce(const float* __restrict__ Do, const float* __restrict__ ws,
                           float* __restrict__ out) {
    int tid = blockIdx.x * blockDim.x + threadIdx.x;
    if (tid >= B_DIM * M_DIM) return;
    int b = tid / M_DIM, m = tid % M_DIM;
    float sr = 0.f, si = 0.f;
    for (int k = 0; k < K_DIM; ++k) {
        int idx = b * (K_DIM * M_DIM) + k * M_DIM + m;
        sr += ws[WS_CSR + idx];
        si += ws[WS_CSI + idx];
    }
    out[2 * tid + 0] = sr * (1.f / K_DIM) + Do[m];
    out[2 * tid + 1] = si * (1.f / K_DIM);
}

extern "C" void kernel_launch(void* const* d_in, const int* in_sizes, int n_in,
                              void* d_out, int out_size, void* d_ws, size_t ws_size,
                              hipStream_t stream) {
    const float* x     = (const float*)d_in[0];
    const float* theta = (const float*)d_in[1];
    const float* lnCr  = (const float*)d_in[2];
    const float* lnCi  = (const float*)d_in[3];
    const float* Dm    = (const float*)d_in[4];
    const float* Do    = (const float*)d_in[5];
    float* ws  = (float*)d_ws;
    float* out = (float*)d_out;

    lds_stage_x<<<(K_DIM * B_DIM * P_DIM + 255) / 256, 256, 0, stream>>>(x, ws);
    lds_lam_bp <<<(K_DIM * N_DIM + 255) / 256, 256, 0, stream>>>(theta, ws);
    lds_state  <<<K_DIM * 4, 32, 0, stream>>>(ws);
    lds_proj   <<<K_DIM * 8, 32, 0, stream>>>(x, lnCr, lnCi, Dm, ws);
    lds_reduce <<<(B_DIM * M_DIM + 255) / 256, 256, 0, stream>>>(Do, ws, out);
}